// DebiasNtXentLoss_39994735461047
// MI455X (gfx1250) — compile-verified
//
#include <hip/hip_runtime.h>
#include <math.h>

// DebiasNtXentLoss, B=4096, D=128, N=8192, T=0.5, rho=0.1
// Fused: normalize -> (WMMA f32 Gram + exp + rowsum, never materialize sim) -> epilogue.

typedef __attribute__((ext_vector_type(2))) float v2f;
typedef __attribute__((ext_vector_type(8))) float v8f;

constexpr int kB   = 4096;
constexpr int kD   = 128;
constexpr int kN   = 8192;
constexpr int kNT  = kN / 16;       // 512 tiles per dimension
constexpr float kInvT = 2.0f;       // 1 / TEMPERATURE
constexpr float kRho  = 0.1f;

__device__ __forceinline__ float wave_reduce_add32(float x) {
  x += __shfl_xor(x, 1);
  x += __shfl_xor(x, 2);
  x += __shfl_xor(x, 4);
  x += __shfl_xor(x, 8);
  x += __shfl_xor(x, 16);
  return x;
}

// --- Kernel 1: row-normalize z = [z_i; z_j] into zn, zero rowsum -------------
__global__ void __launch_bounds__(256)
k_normalize(const float* __restrict__ zi, const float* __restrict__ zj,
            float* __restrict__ zn, float* __restrict__ rowsum) {
  const int lane = threadIdx.x & 31;
  const int wave = threadIdx.x >> 5;
  const int row  = blockIdx.x * 8 + wave;          // 0..N-1, one wave per row
  const float* src = (row < kB) ? (zi + (size_t)row * kD)
                                : (zj + (size_t)(row - kB) * kD);
  float4 v = ((const float4*)src)[lane];           // 32 lanes * 4 = 128
  float ss = v.x * v.x + v.y * v.y + v.z * v.z + v.w * v.w;
  ss = wave_reduce_add32(ss);
  const float inv = 1.0f / fmaxf(sqrtf(ss), 1e-8f);  // torch eps clamp
  float4 o;
  o.x = v.x * inv; o.y = v.y * inv; o.z = v.z * inv; o.w = v.w * inv;
  ((float4*)(zn + (size_t)row * kD))[lane] = o;
  if (lane == 0) rowsum[row] = 0.0f;
}

// --- Kernel 2: positives pos[i] = pos[i+B] = exp(dot(zn_i, zn_{i+B})/T) ------
__global__ void __launch_bounds__(256)
k_pos(const float* __restrict__ zn, float* __restrict__ pos) {
  const int lane = threadIdx.x & 31;
  const int wave = threadIdx.x >> 5;
  const int i    = blockIdx.x * 8 + wave;          // 0..B-1, one wave per pair
  float4 a = ((const float4*)(zn + (size_t)i * kD))[lane];
  float4 b = ((const float4*)(zn + (size_t)(i + kB) * kD))[lane];
  float d = a.x * b.x + a.y * b.y + a.z * b.z + a.w * b.w;
  d = wave_reduce_add32(d);
  if (lane == 0) {
    const float p = __expf(d * kInvT);
    pos[i] = p;
    pos[i + kB] = p;
  }
}

// --- Kernel 3: fused Gram-matrix rowsum via f32 WMMA -------------------------
// Workgroup owns row-tile mt (16 rows). 8 waves split the 512 column tiles.
// A fragment (16x4 f32) lane layout: lane l holds row (l&15), K = 2*(l>>4)+{0,1}
// B fragment (4x16 f32) lane layout: lane l holds col (l&15), K = 2*(l>>4)+{0,1}
// -> identical gather pattern for a Gram matrix; one b64 load per frag per lane.
__global__ void __launch_bounds__(256)
k_gram_rowsum(const float* __restrict__ zn, float* __restrict__ rowsum) {
  const int lane = threadIdx.x & 31;
  const int wave = threadIdx.x >> 5;               // 0..7
  const int mt   = blockIdx.x;                     // 0..511 row tile
  const int r16  = lane & 15;
  const int h2   = (lane >> 4) << 1;               // 0 or 2

  // Hoist all A fragments for this row tile into registers (64 VGPRs).
  const float* arow = zn + (size_t)(mt * 16 + r16) * kD + h2;
  v2f a[32];
#pragma unroll
  for (int k = 0; k < 32; ++k) a[k] = *(const v2f*)(arow + 4 * k);

  v8f acc = {};                                    // per-lane exp-sum accum
  for (int nt = wave; nt < kNT; nt += 8) {
    const float* brow = zn + (size_t)(nt * 16 + r16) * kD + h2;
    v8f c = {};
#pragma unroll
    for (int k = 0; k < 32; ++k) {                 // K = 128 in steps of 4
      v2f b = *(const v2f*)(brow + 4 * k);
      // D = A(16x4) x B(4x16) + C ; full f32 path
      c = __builtin_amdgcn_wmma_f32_16x16x4_f32(
          /*neg_a=*/false, a[k], /*neg_b=*/false, b,
          /*c_mod=*/(short)0, c, /*reuse_a=*/false, /*reuse_b=*/false);
    }
#pragma unroll
    for (int v = 0; v < 8; ++v) acc[v] += __expf(c[v] * kInvT);
  }

  // C/D layout: VGPR v, lanes 0-15 -> row M=v (N=lane), lanes 16-31 -> M=v+8.
  // Row sum = reduce over the 16 lanes of each half, then atomic into rowsum.
#pragma unroll
  for (int v = 0; v < 8; ++v) {
    float s = acc[v];
    s += __shfl_xor(s, 1);
    s += __shfl_xor(s, 2);
    s += __shfl_xor(s, 4);
    s += __shfl_xor(s, 8);
    if ((lane & 15) == 0) {
      const int row = mt * 16 + v + ((lane >> 4) << 3);
      atomicAdd(&rowsum[row], s);
    }
  }
}

// --- Kernel 4: debiased NT-Xent epilogue + mean ------------------------------
__global__ void __launch_bounds__(1024)
k_final(const float* __restrict__ rowsum, const float* __restrict__ pos,
        float* __restrict__ out) {
  __shared__ float red[32];
  const int tid = threadIdx.x;
  const float self_term = __expf(kInvT);           // exp(1/T): zn.zn == 1
  const float n_neg = (float)(kN - 2);
  const float ng_floor = n_neg * __expf(-kInvT);   // (N-2)*exp(-1/T)
  float local = 0.0f;
  for (int i = tid; i < kN; i += 1024) {
    const float p = pos[i];
    const float neg_sum = rowsum[i] - self_term - p;
    float Ng = fmaf(-kRho * n_neg, p, neg_sum) / (1.0f - kRho);
    Ng = fmaxf(Ng, ng_floor);
    local += __logf((p + Ng) / p);                 // -log(p/(p+Ng))
  }
  local = wave_reduce_add32(local);
  if ((tid & 31) == 0) red[tid >> 5] = local;
  __syncthreads();
  if (tid < 32) {
    float s = red[tid];
    s = wave_reduce_add32(s);
    if (tid == 0) out[0] = s / (float)kN;
  }
}

extern "C" void kernel_launch(void* const* d_in, const int* in_sizes, int n_in,
                              void* d_out, int out_size, void* d_ws, size_t ws_size,
                              hipStream_t stream) {
  (void)in_sizes; (void)n_in; (void)out_size; (void)ws_size;
  const float* zi = (const float*)d_in[0];
  const float* zj = (const float*)d_in[1];

  // Workspace layout (floats): zn [N*D] | rowsum [N] | pos [N]  (~4.06 MB)
  float* ws     = (float*)d_ws;
  float* zn     = ws;
  float* rowsum = ws + (size_t)kN * kD;
  float* pos    = rowsum + kN;

  k_normalize  <<<kN / 8, 256, 0, stream>>>(zi, zj, zn, rowsum);
  k_pos        <<<kB / 8, 256, 0, stream>>>(zn, pos);
  k_gram_rowsum<<<kNT,    256, 0, stream>>>(zn, rowsum);
  k_final      <<<1,     1024, 0, stream>>>(rowsum, pos, (float*)d_out);
}